// DigitCapsLayer_38259568673136
// MI455X (gfx1250) — compile-verified
//
#include <hip/hip_runtime.h>
#include <hip/hip_bf16.h>

// Problem constants (from reference setup_inputs)
#define BB   64      // batch
#define DD   10      // digit capsules
#define PP   4096    // primary capsules
#define INSZ 8       // in dim (K)
#define OUTSZ 16     // out dim (M)

typedef __attribute__((ext_vector_type(2))) float v2f;
typedef __attribute__((ext_vector_type(8))) float v8f;

// ---------------------------------------------------------------------------
// Kernel 1: u_hat[b,d,p,o] = sum_i W[d,p,o,i] * x[b,p,i]
// Batched GEMM per (d,p): C[16 o x 64 b] = W[d,p](16x8) @ X_p(8x64)
// via V_WMMA_F32_16X16X4_F32 (M=16,N=16 tile,K=4): 4 b-tiles x 2 K-chunks.
// One wave per (d,p); 8 waves (8 consecutive p, same d) per workgroup.
// Output staged through LDS so global stores are 512B-contiguous per b.
// ---------------------------------------------------------------------------
__global__ __launch_bounds__(256) void uhat_wmma_kernel(
    const float* __restrict__ x,     // [B, P, IN]
    const float* __restrict__ W,     // [D, P, OUT, IN]
    float* __restrict__ uhat)        // [B, D, P, OUT]
{
    // tile[b][wv*16 + o], row stride padded 128 -> 132 floats (bank spread)
    __shared__ float tile[BB * 132];

    const int tid  = threadIdx.x;
    const int lane = tid & 31;
    const int wv   = tid >> 5;          // wave id = local p
    const int d    = blockIdx.y;
    const int p0   = blockIdx.x * 8;
    const int p    = p0 + wv;

    const int oa = lane & 15;           // A row (o) / B col (b in tile)
    const int h  = lane >> 4;           // lane half selects K pair {2h, 2h+1}

    // A fragments from W[d,p,o,i] (i contiguous): float2 at i = 2h (+4 for chunk 2)
    const float* Wp = W + (((size_t)d * PP + p) * OUTSZ + oa) * INSZ;
    v2f A1 = *(const v2f*)(Wp + 2 * h);        // K = 0..3 chunk
    v2f A2 = *(const v2f*)(Wp + 4 + 2 * h);    // K = 4..7 chunk

    #pragma unroll
    for (int t = 0; t < 4; ++t) {               // 4 b-tiles of 16
        const int b = t * 16 + oa;
        const float* xp = x + ((size_t)b * PP + p) * INSZ;
        v2f B1 = *(const v2f*)(xp + 2 * h);
        v2f B2 = *(const v2f*)(xp + 4 + 2 * h);

        v8f acc = {};
        acc = __builtin_amdgcn_wmma_f32_16x16x4_f32(
                  false, A1, false, B1, (short)0, acc, false, false);
        acc = __builtin_amdgcn_wmma_f32_16x16x4_f32(
                  false, A2, false, B2, (short)0, acc, false, false);

        // C layout: lane L, elem r -> (o = r + 8*(L/16), bcol = L%16)
        #pragma unroll
        for (int r = 0; r < 8; ++r) {
            tile[(t * 16 + oa) * 132 + wv * 16 + (r + 8 * h)] = acc[r];
        }
    }
    __syncthreads();

    // Coalesced store: per b, 8 p x 16 o = 128 contiguous floats in [b,d,p,o]
    const size_t dbase = (size_t)d * (PP * OUTSZ) + (size_t)p0 * OUTSZ;
    #pragma unroll
    for (int k = 0; k < 8; ++k) {
        const int i4    = (tid + k * 256) * 4;
        const int b     = i4 >> 7;          // /128
        const int inner = i4 & 127;
        float4 v = *(const float4*)&tile[b * 132 + inner];
        *(float4*)(uhat + (size_t)b * (DD * PP * OUTSZ) + dbase + inner) = v;
    }
}

// ---------------------------------------------------------------------------
// Kernel 2: dynamic routing, one workgroup per (b,d).
// u_hat[b,d,:,:] (256 KB) staged into LDS transposed to [o][p] so every
// P-reduction is stride-1 across lanes (conflict-free). 3 iterations:
//   c = softmax(b_logits over P); s[o] = sum_p c*U; v = squash(s);
//   iters 0,1: b_logits[p] += dot(v, U[:,p]); iter 2: out = v.
// ---------------------------------------------------------------------------
__global__ __launch_bounds__(256) void routing_kernel(
    const float* __restrict__ uhat,  // [B, D, P, OUT]
    float* __restrict__ out)         // [B, D, OUT]
{
    extern __shared__ float smem[];
    float* U    = smem;                   // [16][4096] = 256 KB, U[o*PP + p]
    float* blog = smem + OUTSZ * PP;      // [4096] routing logits
    float* red  = blog + PP;              // [256] reduction scratch

    const int tid  = threadIdx.x;
    const int lane = tid & 31;
    const int wv   = tid >> 5;
    const int bd   = blockIdx.x;          // = b*DD + d (matches uhat layout)

    const float* Ug = uhat + (size_t)bd * (PP * OUTSZ);

    // Load 256 KB slice coalesced (float4) + transpose into [o][p]
    #pragma unroll 4
    for (int k = 0; k < 64; ++k) {
        const int i4 = (tid + k * 256) * 4;    // flat = p*16 + o
        const int p  = i4 >> 4;
        const int o0 = i4 & 15;
        float4 v = *(const float4*)(Ug + i4);
        U[(o0 + 0) * PP + p] = v.x;
        U[(o0 + 1) * PP + p] = v.y;
        U[(o0 + 2) * PP + p] = v.z;
        U[(o0 + 3) * PP + p] = v.w;
    }
    #pragma unroll
    for (int k = 0; k < 16; ++k) blog[k * 256 + tid] = 0.0f;
    __syncthreads();

    float sv[16];
    for (int iter = 0; iter < 3; ++iter) {
        // ---- max over blog (softmax stability) ----
        float m = -3.4e38f;
        #pragma unroll
        for (int k = 0; k < 16; ++k) m = fmaxf(m, blog[k * 256 + tid]);
        #pragma unroll
        for (int off = 16; off > 0; off >>= 1)
            m = fmaxf(m, __shfl_down(m, off, 32));
        if (lane == 0) red[wv] = m;
        __syncthreads();
        float M = red[0];
        #pragma unroll
        for (int w = 1; w < 8; ++w) M = fmaxf(M, red[w]);
        __syncthreads();

        // ---- combined Z and unnormalized s[o] = sum_p e_p * U[o,p] ----
        float part[17];
        #pragma unroll
        for (int j = 0; j < 17; ++j) part[j] = 0.0f;
        for (int k = 0; k < 16; ++k) {
            const int p = k * 256 + tid;
            const float e = __expf(blog[p] - M);
            part[16] += e;
            #pragma unroll
            for (int o = 0; o < 16; ++o) part[o] += e * U[o * PP + p];
        }
        #pragma unroll
        for (int off = 16; off > 0; off >>= 1) {
            #pragma unroll
            for (int j = 0; j < 17; ++j)
                part[j] += __shfl_down(part[j], off, 32);
        }
        if (lane == 0) {
            #pragma unroll
            for (int j = 0; j < 17; ++j) red[16 + wv * 17 + j] = part[j];
        }
        __syncthreads();
        if (tid < 17) {
            float ssum = 0.0f;
            #pragma unroll
            for (int w = 0; w < 8; ++w) ssum += red[16 + w * 17 + tid];
            red[200 + tid] = ssum;
        }
        __syncthreads();

        // ---- squash ----
        const float Z = red[200 + 16];
        float sq = 0.0f;
        #pragma unroll
        for (int o = 0; o < 16; ++o) {
            sv[o] = red[200 + o] / Z;
            sq += sv[o] * sv[o];
        }
        const float scale = (sq / (1.0f + sq)) * rsqrtf(sq + 1e-12f);
        __syncthreads();

        if (iter == 2) {
            if (tid < 16) out[bd * OUTSZ + tid] = scale * sv[tid];
        } else {
            // agreement update: b[p] += sum_o v[o] * U[o,p]
            for (int k = 0; k < 16; ++k) {
                const int p = k * 256 + tid;
                float dot = 0.0f;
                #pragma unroll
                for (int o = 0; o < 16; ++o) dot += sv[o] * U[o * PP + p];
                blog[p] += dot * scale;
            }
            __syncthreads();
        }
    }
}

// ---------------------------------------------------------------------------
extern "C" void kernel_launch(void* const* d_in, const int* in_sizes, int n_in,
                              void* d_out, int out_size, void* d_ws, size_t ws_size,
                              hipStream_t stream) {
    const float* x = (const float*)d_in[0];   // [64, 4096, 8]
    const float* W = (const float*)d_in[1];   // [10, 4096, 16, 8]
    float* out  = (float*)d_out;              // [64, 10, 16]
    float* uhat = (float*)d_ws;               // [64, 10, 4096, 16] = 160 MB scratch

    dim3 g1(PP / 8, DD);                      // one wave per (d,p)
    uhat_wmma_kernel<<<g1, 256, 0, stream>>>(x, W, uhat);

    const size_t smem_bytes = (size_t)(OUTSZ * PP + PP + 256) * sizeof(float); // ~273 KB
    routing_kernel<<<BB * DD, 256, smem_bytes, stream>>>(uhat, out);
}